// BEV2RV_68831145886346
// MI455X (gfx1250) — compile-verified
//
#include <hip/hip_runtime.h>
#include <math.h>

typedef unsigned int v4u __attribute__((ext_vector_type(4)));
typedef int v8i __attribute__((ext_vector_type(8)));
typedef int v4i __attribute__((ext_vector_type(4)));

#define H_B 512
#define W_B 512
#define NCELL (H_B * W_B)            // 262144
#define H_R 64
#define W_R 2048
#define CCH 64
#define NBATCH 2
#define CELLS 128                    // cells per block == TDM tile width (512B rows)

#define PHI_MIN_F  (-3.14159265358979323846f)
#define TWO_PI_F   (6.28318530717958647692f)
#define TH_MAX_F   (0.05235987755982988f)    /* radians(3)   */
#define TH_MIN_F   (-0.4363323129985824f)    /* radians(-25) */
#define DTH_F      (TH_MAX_F - TH_MIN_F)

__device__ __forceinline__ int clampi(int v, int lo, int hi) {
  return v < lo ? lo : (v > hi ? hi : v);
}

// ---------------- pass 2: TDM-staged scatter-max ----------------
// Verified codegen: tensor_load_to_lds (2-group D#), s_wait_tensorcnt,
// global_atomic_max_num_f32 (native, non-returning, L2-resident output).
__global__ __launch_bounds__(CELLS) void bev2rv_scatter(
    const float* __restrict__ bev, const int* __restrict__ zbin,
    float* __restrict__ out) {
  // tile[c*CELLS + cell] = bev[b][c][base + cell]  (32 KB LDS)
  __shared__ float tile[CCH * CELLS];

  const int blocksPerBatch = NCELL / CELLS;
  const int b    = blockIdx.x / blocksPerBatch;
  const int base = (blockIdx.x - b * blocksPerBatch) * CELLS;

  // --- Tensor Data Mover: async 2D tile load, 64 rows (channels) x 128 floats,
  //     row stride = NCELL floats. Issued by wave 0 only (TDM ignores EXEC).
  if (threadIdx.x == 0) {
    unsigned long long gaddr =
        (unsigned long long)(const void*)(bev + (size_t)b * CCH * NCELL + base);
    unsigned int laddr = (unsigned int)(unsigned long long)(const void*)&tile[0];

    v4u g0;
    g0.x = 1u;                                          // count=1 (valid user D#)
    g0.y = laddr;                                       // lds_addr [63:32]
    g0.z = (unsigned int)(gaddr & 0xFFFFFFFFull);       // global_addr [95:64]
    g0.w = (unsigned int)((gaddr >> 32) & 0x1FFFFFFull) // global_addr [120:96]
           | (2u << 30);                                // type=2 ("image")

    v8i g1;
    g1[0] = 0x00020000;                                 // data_size = 4B
    g1[1] = (int)((NCELL & 0xFFFF) << 16);              // tensor_dim0 lo16
    g1[2] = (int)(((NCELL >> 16) & 0xFFFF)              // tensor_dim0 hi16
           | (CCH << 16));                              // tensor_dim1 lo16 = 64
    g1[3] = (int)(CELLS << 16);                         // tile_dim0 = 128
    g1[4] = CCH;                                        // tile_dim1 = 64
    g1[5] = NCELL;                                      // tensor_dim0_stride lo32
    g1[6] = 0;
    g1[7] = 0;

    v4i g2 = {0, 0, 0, 0};
    v4i g3 = {0, 0, 0, 0};
#if defined(__clang_major__) && (__clang_major__ >= 23)
    v8i g4 = {0, 0, 0, 0, 0, 0, 0, 0};
    __builtin_amdgcn_tensor_load_to_lds(g0, g1, g2, g3, g4, 0);
#else
    __builtin_amdgcn_tensor_load_to_lds(g0, g1, g2, g3, 0);
#endif
  }

  // --- per-cell geometry overlapped with the in-flight DMA ---
  const int cell = threadIdx.x;
  const int n = base + cell;
  const int zb = zbin[b * NCELL + n];          // issue early

  const int i = n >> 9;            // BEV row
  const int j = n & (W_B - 1);     // BEV col
  const float step = 100.0f / 511.0f;
  const float x = -50.0f + step * (float)j;
  const float y =  50.0f - step * (float)i;
  const float rho = sqrtf(x * x + y * y);
  const float phi = atan2f(y, x);

  const int col = clampi((int)rintf((phi - PHI_MIN_F) / TWO_PI_F * (float)(W_R - 1)),
                         0, W_R - 1);

  const float th_low = atan2f(-1.73f, rho);
  const int row_low = clampi((int)rintf((TH_MAX_F - th_low) / DTH_F * (float)(H_R - 1)),
                             0, H_R - 1);

  const float z_hint = 0.2f * (float)zb - 3.9f;        // zb*DZ + (Z_MIN + DZ/2)
  const float th_hi = atan2f(z_hint, rho);
  const int row_hi = clampi((int)rintf((TH_MAX_F - th_hi) / DTH_F * (float)(H_R - 1)),
                            0, H_R - 1);

  const int rs = row_low < row_hi ? row_low : row_hi;
  const int re = row_low < row_hi ? row_hi : row_low;

  // wave 0 drains its TENSORcnt, then everyone syncs before reading the tile
  if (threadIdx.x == 0) {
    __builtin_amdgcn_s_wait_tensorcnt(0);
  }
  __syncthreads();

  float* __restrict__ outB = out + (size_t)b * CCH * H_R * W_R;
  for (int c = 0; c < CCH; ++c) {
    const float v = tile[c * CELLS + cell];
    float* p = outB + ((size_t)c * H_R + rs) * W_R + col;
    for (int r = rs; r <= re; ++r) {
      // IEEE maximumNumber f32 atomic (global_atomic_max_num_f32)
      (void)__hip_atomic_fetch_max(p, v, __ATOMIC_RELAXED, __HIP_MEMORY_SCOPE_AGENT);
      p += W_R;
    }
  }
}

// ---------------- pass 1: init output to -inf ----------------
__global__ void bev2rv_init(float4* __restrict__ out, int n4) {
  int i = blockIdx.x * blockDim.x + threadIdx.x;
  if (i < n4) out[i] = make_float4(-INFINITY, -INFINITY, -INFINITY, -INFINITY);
}

// ---------------- pass 3: -inf -> 0 ----------------
__global__ void bev2rv_finalize(float4* __restrict__ out, int n4) {
  int i = blockIdx.x * blockDim.x + threadIdx.x;
  if (i < n4) {
    float4 v = out[i];
    v.x = (v.x == -INFINITY) ? 0.0f : v.x;
    v.y = (v.y == -INFINITY) ? 0.0f : v.y;
    v.z = (v.z == -INFINITY) ? 0.0f : v.z;
    v.w = (v.w == -INFINITY) ? 0.0f : v.w;
    out[i] = v;
  }
}

extern "C" void kernel_launch(void* const* d_in, const int* in_sizes, int n_in,
                              void* d_out, int out_size, void* d_ws, size_t ws_size,
                              hipStream_t stream) {
  const float* bev = (const float*)d_in[0];   // [2, 64, 512, 512] f32
  const int*   zb  = (const int*)d_in[1];     // [2, 1, 512, 512] i32
  float* out = (float*)d_out;                 // [2, 64, 64, 2048] f32

  const int total = NBATCH * CCH * H_R * W_R; // 16,777,216
  const int n4 = total / 4;

  bev2rv_init<<<(n4 + 255) / 256, 256, 0, stream>>>((float4*)out, n4);
  bev2rv_scatter<<<NBATCH * (NCELL / CELLS), CELLS, 0, stream>>>(bev, zb, out);
  bev2rv_finalize<<<(n4 + 255) / 256, 256, 0, stream>>>((float4*)out, n4);
}